// FEModule_8761733284507
// MI455X (gfx1250) — compile-verified
//
#include <hip/hip_runtime.h>
#include <hip/hip_bf16.h>

typedef float v2f __attribute__((ext_vector_type(2)));
typedef float v4f __attribute__((ext_vector_type(4)));
typedef float v8f __attribute__((ext_vector_type(8)));
typedef int   i32x4 __attribute__((vector_size(16)));

#define BB   4
#define NN   4096
#define KK   16
#define CC   128
#define C0   259          // 3 + 128 + 128
#define C0P  260          // padded to multiple of 4
#define XP   132          // LDS pitch for weight tile (floats): stride-4 bank walk
#define PS   288          // pair-row pitch for x tiles: 288 % 64 == 32 -> halves split banks
#define TOT  (BB * NN * KK)   // 262144 columns
#define EPSV 1e-5f

// x-tile LDS layout: element (k, c) at ((k>>1)*PS + 2c + (k&1))
// => B fragment (x[ka][c], x[ka+1][c]) is one aligned ds_load_b64.
#define XIDX(r, c) ((((r) >> 1) * PS) + (((c)) << 1) + ((r) & 1))

#define GLOBAL_AS __attribute__((address_space(1)))
#define SHARED_AS __attribute__((address_space(3)))

// Async global->LDS (ASYNCcnt path), guarded so host pass / older toolchains fall back.
#if defined(__has_builtin)
#  if __has_builtin(__builtin_amdgcn_global_load_async_to_lds_b128) && \
      __has_builtin(__builtin_amdgcn_s_wait_asynccnt)
#    define HAVE_ASYNC_LDS 1
#  endif
#endif
#ifndef HAVE_ASYNC_LDS
#  define HAVE_ASYNC_LDS 0
#endif

// ---------------------------------------------------------------------------
// Workspace layout (floats):
//   [0,768)            : stats  (3 layers x 128 ch x {sum, sumsq})
//   [768,1536)         : scale/shift (3 layers x 128 ch x {scale, shift})
//   [1536, +33280)     : W0 padded to 128 x 260
//   idx                : B*N*K ints
//   actA, actB         : 128 x 262144 floats each
// ---------------------------------------------------------------------------
#define WS_WPAD_OFF  1536
#define WS_IDX_OFF   (WS_WPAD_OFF + CC * C0P)
#define WS_ACTA_OFF  (WS_IDX_OFF + TOT)
#define WS_ACTB_OFF  (WS_ACTA_OFF + CC * TOT)

__global__ __launch_bounds__(256) void zero_stats_kernel(float* __restrict__ st) {
    int i = blockIdx.x * 256 + threadIdx.x;
    if (i < 1536) st[i] = 0.0f;
}

__global__ __launch_bounds__(256) void pad_w0_kernel(const float* __restrict__ W0,
                                                     float* __restrict__ Wp) {
    int i = blockIdx.x * 256 + threadIdx.x;
    if (i < CC * C0P) {
        int r = i / C0P, k = i - r * C0P;
        Wp[i] = (k < C0) ? W0[r * C0 + k] : 0.0f;
    }
}

// --------------------------- kNN (top-16) ----------------------------------
__global__ __launch_bounds__(256) void knn_kernel(const float* __restrict__ pos1,
                                                  const float* __restrict__ pos2,
                                                  int* __restrict__ idxout) {
    __shared__ float p2x[256], p2y[256], p2z[256];
    const int tid = threadIdx.x;
    const int b   = blockIdx.x / (NN / 256);
    const int n   = (blockIdx.x % (NN / 256)) * 256 + tid;

    const float px = pos1[(b * 3 + 0) * NN + n];
    const float py = pos1[(b * 3 + 1) * NN + n];
    const float pz = pos1[(b * 3 + 2) * NN + n];

    float bd[KK]; int bi[KK];
#pragma unroll
    for (int j = 0; j < KK; ++j) { bd[j] = 3.4e38f; bi[j] = 0; }
    float worst = 3.4e38f; int wslot = 0;

    for (int tile = 0; tile < NN; tile += 256) {
        __syncthreads();
        p2x[tid] = pos2[(b * 3 + 0) * NN + tile + tid];
        p2y[tid] = pos2[(b * 3 + 1) * NN + tile + tid];
        p2z[tid] = pos2[(b * 3 + 2) * NN + tile + tid];
        __syncthreads();
        for (int i = 0; i < 256; ++i) {
            float dx = px - p2x[i], dy = py - p2y[i], dz = pz - p2z[i];
            float d = dx * dx + dy * dy + dz * dz;
            if (d < worst) {
                int cand = tile + i;
#pragma unroll
                for (int j = 0; j < KK; ++j)
                    if (j == wslot) { bd[j] = d; bi[j] = cand; }
                worst = bd[0]; wslot = 0;
#pragma unroll
                for (int j = 1; j < KK; ++j)
                    if (bd[j] > worst) { worst = bd[j]; wslot = j; }
            }
        }
    }
#pragma unroll
    for (int j = 0; j < KK; ++j)
        idxout[((size_t)b * NN + n) * KK + j] = bi[j];
}

// ------------- Layer 0: fused gather/concat + GEMM (Wp: 128x260) -----------
__global__ __launch_bounds__(256) void gather_gemm0_kernel(
        const float* __restrict__ pos1, const float* __restrict__ pos2,
        const float* __restrict__ feat1, const float* __restrict__ feat2,
        const float* __restrict__ Wp, const int* __restrict__ knn,
        float* __restrict__ yout) {
    __shared__ float xs[(C0P / 2) * PS];    // pair-interleaved 260 x 128 tile
    __shared__ int   jcol[128];
    const int tid     = threadIdx.x;
    const int colbase = blockIdx.x * 128;
    const int b       = colbase / (NN * KK);
    const int nbase   = (colbase % (NN * KK)) / KK;   // 8 points per WG

    // Phase 1: gather the 259x128 x-tile into LDS
    if (tid < 128) {
        int c = tid;
        int n = nbase + (c >> 4);
        int k = c & 15;
        int j = knn[((size_t)b * NN + n) * KK + k];
        jcol[c] = j;
#pragma unroll
        for (int r = 0; r < 3; ++r)
            xs[XIDX(r, c)] = pos2[(b * 3 + r) * NN + j] - pos1[(b * 3 + r) * NN + n];
        xs[XIDX(259, c)] = 0.0f;   // padding row
    }
    __syncthreads();
    for (int e = tid; e < 128 * 128; e += 256) {       // gathered feature2
        int r = e >> 7, c = e & 127;
        xs[XIDX(3 + r, c)] = feat2[((size_t)b * CC + r) * NN + jcol[c]];
    }
    for (int e = tid; e < 128 * 128; e += 256) {       // broadcast feature1
        int r = e >> 7, c = e & 127;
        int n = nbase + (c >> 4);
        xs[XIDX(131 + r, c)] = feat1[((size_t)b * CC + r) * NN + n];
    }
    __syncthreads();

    // Phase 2: WMMA fp32 16x16x4.  Wave w owns output rows [16w,16w+16).
    const int wave = tid >> 5;
    const int lane = tid & 31;
    const int half = lane >> 4;
    const int l16  = lane & 15;
    const int mrow = wave * 16 + l16;

    v8f acc[8] = {};
    for (int k0 = 0; k0 < C0P; k0 += 4) {
        const int ka = k0 + half * 2;
        const int kp = ka >> 1;                        // pair row
        v2f a = *(const v2f*)(Wp + mrow * C0P + ka);   // unguarded, 8B aligned
#pragma unroll
        for (int t = 0; t < 8; ++t) {
            v2f bm = *(const v2f*)&xs[kp * PS + ((t * 16 + l16) << 1)];
            acc[t] = __builtin_amdgcn_wmma_f32_16x16x4_f32(
                false, a, false, bm, (short)0, acc[t], false, false);
        }
    }
#pragma unroll
    for (int t = 0; t < 8; ++t) {
        int col = colbase + t * 16 + l16;
#pragma unroll
        for (int j = 0; j < 8; ++j) {
            int m = wave * 16 + j + 8 * half;
            yout[(size_t)m * TOT + col] = acc[t][j];
        }
    }
}

// ------------- Layers 1/2: BN+ReLU fused into GEMM input (W: 128x128) ------
// W is staged into LDS via async global->LDS copies overlapped with the
// BN+ReLU x-tile load; fenced with s_wait_asynccnt 0 + workgroup barrier.
__global__ __launch_bounds__(256) void bn_gemm_kernel(
        const float* __restrict__ yin, const float* __restrict__ ss,
        const float* __restrict__ W, float* __restrict__ yout) {
    __shared__ float xs[(CC / 2) * PS];     // pair-interleaved 128 x 128 tile
    __shared__ float wl[CC * XP];
    const int tid     = threadIdx.x;
    const int colbase = blockIdx.x * 128;

    // stage W (128x128 -> pitch-132 LDS), 4096 x b128 chunks
#if HAVE_ASYNC_LDS
    for (int e = tid; e < CC * 32; e += 256) {
        int r = e >> 5, c4 = (e & 31) << 2;
        __builtin_amdgcn_global_load_async_to_lds_b128(
            (GLOBAL_AS i32x4*)(W + r * CC + c4),
            (SHARED_AS i32x4*)&wl[r * XP + c4], 0, 0);
    }
#else
    for (int e = tid; e < CC * 32; e += 256) {
        int r = e >> 5, c4 = (e & 31) << 2;
        *(v4f*)&wl[r * XP + c4] = *(const v4f*)(W + r * CC + c4);
    }
#endif

    // x tile with BN + ReLU applied on load
    for (int e = tid; e < 128 * 128; e += 256) {
        int r = e >> 7, c = e & 127;
        float v = yin[(size_t)r * TOT + colbase + c];
        xs[XIDX(r, c)] = fmaxf(v * ss[2 * r] + ss[2 * r + 1], 0.0f);
    }
#if HAVE_ASYNC_LDS
    __builtin_amdgcn_s_wait_asynccnt(0);
#endif
    __syncthreads();

    const int wave = tid >> 5;
    const int lane = tid & 31;
    const int half = lane >> 4;
    const int l16  = lane & 15;
    const int mrow = wave * 16 + l16;

    v8f acc[8] = {};
    for (int k0 = 0; k0 < CC; k0 += 4) {
        const int ka = k0 + half * 2;
        const int kp = ka >> 1;
        v2f a = *(const v2f*)&wl[mrow * XP + ka];      // LDS A-fragment
#pragma unroll
        for (int t = 0; t < 8; ++t) {
            v2f bm = *(const v2f*)&xs[kp * PS + ((t * 16 + l16) << 1)];
            acc[t] = __builtin_amdgcn_wmma_f32_16x16x4_f32(
                false, a, false, bm, (short)0, acc[t], false, false);
        }
    }
#pragma unroll
    for (int t = 0; t < 8; ++t) {
        int col = colbase + t * 16 + l16;
#pragma unroll
        for (int j = 0; j < 8; ++j) {
            int m = wave * 16 + j + 8 * half;
            yout[(size_t)m * TOT + col] = acc[t][j];
        }
    }
}

// --------------------- per-channel sum / sumsq ------------------------------
__global__ __launch_bounds__(256) void stats_kernel(const float* __restrict__ y,
                                                    float* __restrict__ st) {
    const int ch      = blockIdx.y;
    const size_t base = (size_t)ch * TOT + (size_t)blockIdx.x * 8192;
    float s = 0.0f, s2 = 0.0f;
    for (int i = threadIdx.x; i < 8192; i += 256) {
        float v = y[base + i];
        s += v; s2 += v * v;
    }
    for (int off = 16; off > 0; off >>= 1) {
        s  += __shfl_down(s,  off, 32);
        s2 += __shfl_down(s2, off, 32);
    }
    __shared__ float ls[8], ls2[8];
    const int w = threadIdx.x >> 5, l = threadIdx.x & 31;
    if (l == 0) { ls[w] = s; ls2[w] = s2; }
    __syncthreads();
    if (threadIdx.x == 0) {
        float t = 0.0f, t2 = 0.0f;
#pragma unroll
        for (int i = 0; i < 8; ++i) { t += ls[i]; t2 += ls2[i]; }
        atomicAdd(&st[2 * ch],     t);
        atomicAdd(&st[2 * ch + 1], t2);
    }
}

__global__ void finalize_kernel(const float* __restrict__ st,
                                const float* __restrict__ g,
                                const float* __restrict__ bt,
                                float* __restrict__ ss) {
    int c = threadIdx.x;
    if (c >= CC) return;
    const float inv  = 1.0f / (float)TOT;
    float mean = st[2 * c] * inv;
    float var  = st[2 * c + 1] * inv - mean * mean;
    float sc   = g[c] * rsqrtf(var + EPSV);
    ss[2 * c]     = sc;
    ss[2 * c + 1] = bt[c] - mean * sc;
}

// --------------------- final BN + ReLU + max over K -------------------------
__global__ __launch_bounds__(256) void bn_max_kernel(const float* __restrict__ y,
                                                     const float* __restrict__ ss,
                                                     float* __restrict__ out) {
    int gid = blockIdx.x * 256 + threadIdx.x;    // over B*C*N
    if (gid >= BB * CC * NN) return;
    int n = gid % NN;
    int c = (gid / NN) % CC;
    int b = gid / (NN * CC);
    size_t base = (size_t)c * TOT + ((size_t)b * NN + n) * KK;
    float sc = ss[2 * c], sh = ss[2 * c + 1];
    float m = 0.0f;                              // ReLU outputs are >= 0
#pragma unroll
    for (int k = 0; k < KK; ++k)
        m = fmaxf(m, fmaxf(y[base + k] * sc + sh, 0.0f));
    out[gid] = m;
}

__global__ __launch_bounds__(256) void copy_pos_kernel(const float* __restrict__ p,
                                                       float* __restrict__ out) {
    int i = blockIdx.x * 256 + threadIdx.x;
    if (i < BB * 3 * NN) out[i] = p[i];
}

// ---------------------------------------------------------------------------
extern "C" void kernel_launch(void* const* d_in, const int* in_sizes, int n_in,
                              void* d_out, int out_size, void* d_ws, size_t ws_size,
                              hipStream_t stream) {
    const float* pos1  = (const float*)d_in[0];
    const float* pos2  = (const float*)d_in[1];
    const float* feat1 = (const float*)d_in[2];
    const float* feat2 = (const float*)d_in[3];
    const float* W0    = (const float*)d_in[4];
    const float* W1    = (const float*)d_in[5];
    const float* W2    = (const float*)d_in[6];
    const float* g0    = (const float*)d_in[7];
    const float* g1    = (const float*)d_in[8];
    const float* g2    = (const float*)d_in[9];
    const float* b0    = (const float*)d_in[10];
    const float* b1    = (const float*)d_in[11];
    const float* b2    = (const float*)d_in[12];

    float* ws    = (float*)d_ws;
    float* stats = ws;                    // 3 x 256
    float* ss    = ws + 768;              // 3 x 256
    float* Wp    = ws + WS_WPAD_OFF;      // 128 x 260
    int*   idx   = (int*)(ws + WS_IDX_OFF);
    float* actA  = ws + WS_ACTA_OFF;
    float* actB  = ws + WS_ACTB_OFF;
    float* outp  = (float*)d_out;

    const dim3 blk(256);

    zero_stats_kernel<<<6, blk, 0, stream>>>(stats);
    pad_w0_kernel<<<(CC * C0P + 255) / 256, blk, 0, stream>>>(W0, Wp);
    knn_kernel<<<BB * NN / 256, blk, 0, stream>>>(pos1, pos2, idx);

    // layer 0
    gather_gemm0_kernel<<<TOT / 128, blk, 0, stream>>>(pos1, pos2, feat1, feat2,
                                                       Wp, idx, actA);
    stats_kernel<<<dim3(TOT / 8192, CC), blk, 0, stream>>>(actA, stats + 0);
    finalize_kernel<<<1, 128, 0, stream>>>(stats + 0, g0, b0, ss + 0);

    // layer 1
    bn_gemm_kernel<<<TOT / 128, blk, 0, stream>>>(actA, ss + 0, W1, actB);
    stats_kernel<<<dim3(TOT / 8192, CC), blk, 0, stream>>>(actB, stats + 256);
    finalize_kernel<<<1, 128, 0, stream>>>(stats + 256, g1, b1, ss + 256);

    // layer 2
    bn_gemm_kernel<<<TOT / 128, blk, 0, stream>>>(actB, ss + 256, W2, actA);
    stats_kernel<<<dim3(TOT / 8192, CC), blk, 0, stream>>>(actA, stats + 512);
    finalize_kernel<<<1, 128, 0, stream>>>(stats + 512, g2, b2, ss + 512);

    // outputs: pos1 passthrough + BN/ReLU/max-over-K
    copy_pos_kernel<<<(BB * 3 * NN + 255) / 256, blk, 0, stream>>>(pos1, outp);
    bn_max_kernel<<<(BB * CC * NN + 255) / 256, blk, 0, stream>>>(
        actA, ss + 512, outp + BB * 3 * NN);
}